// TopKTokenChoiceRouterLoRa_65481071411003
// MI455X (gfx1250) — compile-verified
//
#include <hip/hip_runtime.h>
#include <math.h>

#define T_TOK 16384
#define H_DIM 2048
#define E_EXP 64
#define TOPK  2
#define AUX_COEFF 0.01f
#define Z_COEFF   0.001f

#define KC 128                       // K-chunk of W staged in LDS (128*64*4B = 32KB)
#define WAVES_PER_BLOCK 4            // 128 threads -> 256 blocks -> all WGPs busy
#define BLOCK_THREADS (WAVES_PER_BLOCK * 32)
#define TOKENS_PER_WAVE 16
#define TOKENS_PER_BLOCK (WAVES_PER_BLOCK * TOKENS_PER_WAVE)

typedef float v2f __attribute__((ext_vector_type(2)));
typedef float v8f __attribute__((ext_vector_type(8)));

// Types matching the async-to-LDS builtin prototype: V4i pointers in AS(1)/AS(3)
typedef int v4i_g __attribute__((vector_size(4 * sizeof(int))));
typedef __attribute__((address_space(1))) v4i_g glb_v4i;
typedef __attribute__((address_space(3))) v4i_g lds_v4i;

#if __has_builtin(__builtin_amdgcn_global_load_async_to_lds_b128)
#define USE_ASYNC_LDS 1
#endif

// workspace layout (floats): [0] sum(lse^2), [1..64] sum probs per expert, [65..128] routed counts
#define WS_FLOATS 129

__global__ void router_init_ws(float* __restrict__ ws) {
    int i = blockIdx.x * blockDim.x + threadIdx.x;
    if (i < WS_FLOATS) ws[i] = 0.0f;
}

#define WMMA_F32_16x16x4(ACC, A, B)                                            \
    ACC = __builtin_amdgcn_wmma_f32_16x16x4_f32(false, (A), false, (B),        \
                                                (short)0, (ACC), false, false)

__global__ __launch_bounds__(BLOCK_THREADS)
void router_main(const float* __restrict__ x, const float* __restrict__ w,
                 float* __restrict__ out, float* __restrict__ ws) {
    __shared__ float sB[KC * E_EXP];
    __shared__ float sProb[E_EXP];
    __shared__ float sCnt[E_EXP];
    __shared__ float sLse2;

    const int tid  = threadIdx.x;
    const int wave = tid >> 5;
    const int lane = tid & 31;
    const int l16  = lane & 15;
    const int hi   = lane >> 4;   // 0: K-offsets 0,1 ; 1: K-offsets 2,3
    const int rowBase = (blockIdx.x * WAVES_PER_BLOCK + wave) * TOKENS_PER_WAVE;
    const int myRow   = rowBase + l16;

    if (tid < E_EXP) { sProb[tid] = 0.0f; sCnt[tid] = 0.0f; }
    if (tid == 0)    sLse2 = 0.0f;

    v8f acc0 = {}, acc1 = {}, acc2 = {}, acc3 = {};

    const float* xrow = x + (size_t)myRow * H_DIM + (hi << 1);

    for (int kc = 0; kc < H_DIM; kc += KC) {
        __syncthreads();   // protect sB reuse (and cover shared init on iter 0)

        // ---- stage W[kc:kc+KC, :] into LDS ----
#ifdef USE_ASYNC_LDS
        {
            const float* gbase = w + (size_t)kc * E_EXP;
            #pragma unroll
            for (int j = 0; j < (KC * E_EXP / 4) / BLOCK_THREADS; ++j) {
                int off4 = j * BLOCK_THREADS + tid;        // float4 index
                __builtin_amdgcn_global_load_async_to_lds_b128(
                    (glb_v4i*)(unsigned long long)(gbase + off4 * 4),
                    (lds_v4i*)(unsigned int)(unsigned long long)(void*)&sB[off4 * 4],
                    0, 0);
            }
#if __has_builtin(__builtin_amdgcn_s_wait_asynccnt)
            __builtin_amdgcn_s_wait_asynccnt(0);
#else
            asm volatile("s_wait_asynccnt 0" ::: "memory");
#endif
        }
#else
        {
            const float4* gbase = (const float4*)(w + (size_t)kc * E_EXP);
            #pragma unroll
            for (int j = 0; j < (KC * E_EXP / 4) / BLOCK_THREADS; ++j) {
                int off4 = j * BLOCK_THREADS + tid;
                ((float4*)sB)[off4] = gbase[off4];
            }
        }
#endif
        __syncthreads();

        // ---- WMMA inner loop: D(16x16) += A(16x4) x B(4x16), 4 expert tiles ----
        const float* xg = xrow + kc;
        #pragma unroll 4
        for (int kk = 0; kk < KC; kk += 4) {
            const float2 av = *(const float2*)(xg + kk);
            v2f a; a.x = av.x; a.y = av.y;

            const int kB = kk + (hi << 1);
            const float* brow0 = &sB[kB * E_EXP + l16];
            const float* brow1 = brow0 + E_EXP;

            v2f b;
            b.x = brow0[0];  b.y = brow1[0];  WMMA_F32_16x16x4(acc0, a, b);
            b.x = brow0[16]; b.y = brow1[16]; WMMA_F32_16x16x4(acc1, a, b);
            b.x = brow0[32]; b.y = brow1[32]; WMMA_F32_16x16x4(acc2, a, b);
            b.x = brow0[48]; b.y = brow1[48]; WMMA_F32_16x16x4(acc3, a, b);
        }
    }

    // ---- epilogue: token M's 64 logits live on one 16-lane half-group,
    //      VGPR slot v = M % 8, half-group hi = M / 8 ----
    v8f lm, ls, inv;
    #pragma unroll
    for (int v = 0; v < 8; ++v)
        lm[v] = fmaxf(fmaxf(acc0[v], acc1[v]), fmaxf(acc2[v], acc3[v]));
    #pragma unroll
    for (int m = 1; m < 16; m <<= 1) {
        #pragma unroll
        for (int v = 0; v < 8; ++v)
            lm[v] = fmaxf(lm[v], __shfl_xor(lm[v], m, 16));
    }
    #pragma unroll
    for (int v = 0; v < 8; ++v)
        ls[v] = __expf(acc0[v] - lm[v]) + __expf(acc1[v] - lm[v]) +
                __expf(acc2[v] - lm[v]) + __expf(acc3[v] - lm[v]);
    #pragma unroll
    for (int m = 1; m < 16; m <<= 1) {
        #pragma unroll
        for (int v = 0; v < 8; ++v)
            ls[v] += __shfl_xor(ls[v], m, 16);
    }
    #pragma unroll
    for (int v = 0; v < 8; ++v)
        inv[v] = 1.0f / ls[v];

    // per-expert mean-prob partial sums (this lane owns experts t*16 + l16)
    {
        float ps0 = 0.f, ps1 = 0.f, ps2 = 0.f, ps3 = 0.f;
        #pragma unroll
        for (int v = 0; v < 8; ++v) {
            ps0 += __expf(acc0[v] - lm[v]) * inv[v];
            ps1 += __expf(acc1[v] - lm[v]) * inv[v];
            ps2 += __expf(acc2[v] - lm[v]) * inv[v];
            ps3 += __expf(acc3[v] - lm[v]) * inv[v];
        }
        atomicAdd(&sProb[ 0 + l16], ps0);
        atomicAdd(&sProb[16 + l16], ps1);
        atomicAdd(&sProb[32 + l16], ps2);
        atomicAdd(&sProb[48 + l16], ps3);
    }

    // z-loss partial: sum of lse^2 over this half-group's 8 tokens
    if (l16 == 0) {
        float zacc = 0.f;
        #pragma unroll
        for (int v = 0; v < 8; ++v) {
            float lse = lm[v] + __logf(ls[v]);
            zacc += lse * lse;
        }
        atomicAdd(&sLse2, zacc);
    }

    // top-2 per token (reduce over the 16-lane half-group)
    #pragma unroll
    for (int v = 0; v < 8; ++v) {
        float c0 = acc0[v], c1 = acc1[v], c2 = acc2[v], c3 = acc3[v];
        float b1 = c0; int j1 = l16;
        float b2 = c1; int j2 = 16 + l16;
        if (b2 > b1) { float tf = b1; b1 = b2; b2 = tf; int tj = j1; j1 = j2; j2 = tj; }
        if (c2 > b1)      { b2 = b1; j2 = j1; b1 = c2; j1 = 32 + l16; }
        else if (c2 > b2) { b2 = c2; j2 = 32 + l16; }
        if (c3 > b1)      { b2 = b1; j2 = j1; b1 = c3; j1 = 48 + l16; }
        else if (c3 > b2) { b2 = c3; j2 = 48 + l16; }

        #pragma unroll
        for (int m = 1; m < 16; m <<= 1) {
            float o1 = __shfl_xor(b1, m, 16); int oj1 = __shfl_xor(j1, m, 16);
            float o2 = __shfl_xor(b2, m, 16); int oj2 = __shfl_xor(j2, m, 16);
            if (o1 > b1) {
                if (b1 > o2) { b2 = b1; j2 = j1; }
                else         { b2 = o2; j2 = oj2; }
                b1 = o1; j1 = oj1;
            } else if (o1 > b2) {
                b2 = o1; j2 = oj1;
            }
        }

        if (l16 == 0) {
            int tok = rowBase + (hi << 3) + v;
            out[tok * 2 + 0] = __expf(b1 - lm[v]) * inv[v];
            out[tok * 2 + 1] = __expf(b2 - lm[v]) * inv[v];
            out[2 * T_TOK + tok * 2 + 0] = (float)j1;
            out[2 * T_TOK + tok * 2 + 1] = (float)j2;
            atomicAdd(&sCnt[j1], 1.0f);
            atomicAdd(&sCnt[j2], 1.0f);
        }
    }

    __syncthreads();
    if (tid < E_EXP) {
        atomicAdd(&ws[1 + tid],  sProb[tid]);
        atomicAdd(&ws[65 + tid], sCnt[tid]);
    }
    if (tid == 0) atomicAdd(&ws[0], sLse2);
}

__global__ void router_finalize(const float* __restrict__ ws, float* __restrict__ out) {
    if (threadIdx.x == 0 && blockIdx.x == 0) {
        float s = 0.f;
        const float invT = 1.0f / (float)T_TOK;
        for (int e = 0; e < E_EXP; ++e) {
            float meanp = ws[1 + e]  * invT;
            float frac  = ws[65 + e] * invT;
            s += frac * meanp;
        }
        out[4 * T_TOK + 0] = AUX_COEFF * ((float)E_EXP / (float)TOPK) * s;  // aux_loss
        out[4 * T_TOK + 1] = Z_COEFF * (ws[0] * invT);                      // z_loss
    }
}

extern "C" void kernel_launch(void* const* d_in, const int* in_sizes, int n_in,
                              void* d_out, int out_size, void* d_ws, size_t ws_size,
                              hipStream_t stream) {
    (void)in_sizes; (void)n_in; (void)out_size; (void)ws_size;
    const float* x = (const float*)d_in[0];
    const float* w = (const float*)d_in[1];
    float* out = (float*)d_out;
    float* ws  = (float*)d_ws;

    router_init_ws<<<1, 256, 0, stream>>>(ws);
    router_main<<<T_TOK / TOKENS_PER_BLOCK, BLOCK_THREADS, 0, stream>>>(x, w, out, ws);
    router_finalize<<<1, 64, 0, stream>>>(ws, out);
}